// PTABlockWithLog_89747636617463
// MI455X (gfx1250) — compile-verified
//
#include <hip/hip_runtime.h>

typedef float v2f  __attribute__((ext_vector_type(2)));
typedef float v4f  __attribute__((ext_vector_type(4)));
typedef float v8f  __attribute__((ext_vector_type(8)));

#define EPS 1e-10f

// One wave (32 lanes) computes a 16-row tile of the output.
//   A (16x4 f32)  = log2(|x*iw| + eps) chunk       (per ISA layout: lane l
//                    holds rows l&15, K = 2*(l>>4) and K+1 -> one float2 load)
//   B (4x16 f32)  = ln_w chunk broadcast across the 16 columns
//   D (16x16 f32) = row-dot replicated in every column
// out[row] = exp2(D[row, any])   (ln/exp folded into log2/exp2 identities)
__global__ __launch_bounds__(256)
void pta_log_wmma_kernel(const float* __restrict__ x,
                         const float* __restrict__ idw,
                         const float* __restrict__ lnw,
                         float* __restrict__ out,
                         int ntiles, long nrows) {
  const int lane = threadIdx.x & 31;
  const int wave = (blockIdx.x << 3) | (threadIdx.x >> 5);
  if (wave >= ntiles) return;              // wave-uniform: EXEC stays all-1s

  const int  h  = lane >> 4;               // half-wave select (K pair)
  const int  li = lane & 15;               // row within tile / column id

  const long rowBase = (long)wave * 16;
  long row = rowBase + li;
  if (row >= nrows) row = nrows - 1;       // clamp loads for tail tile
  const float* xrow = x + (row << 6);      // F == 64

  // Stream-ahead prefetch: each lane pulls one 128B line of the tile 64
  // tiles (256 KB) downstream -> global_prefetch_b8.
  {
    const long pfRow = rowBase + 16L * 64L;
    if (pfRow + 16 <= nrows)
      __builtin_prefetch(x + (pfRow << 6) + ((long)lane << 5), 0, 0);
  }

  v8f acc = {0.f, 0.f, 0.f, 0.f, 0.f, 0.f, 0.f, 0.f};

#pragma unroll
  for (int kc = 0; kc < 16; ++kc) {
    const int k = (kc << 2) + (h << 1);            // feature pair for this lane
    const v2f xv = *(const v2f*)(xrow + k);        // global_load_b64 (dense)
    const v2f wv = *(const v2f*)(idw  + k);        // cached (512B total)
    const v2f lv = *(const v2f*)(lnw  + k);        // cached (512B total)
    v2f a;
    a.x = __builtin_amdgcn_logf(__builtin_fabsf(xv.x * wv.x) + EPS);
    a.y = __builtin_amdgcn_logf(__builtin_fabsf(xv.y * wv.y) + EPS);
    // D = A x B + C  (v_wmma_f32_16x16x4_f32)
    acc = __builtin_amdgcn_wmma_f32_16x16x4_f32(
        /*neg_a=*/false, a, /*neg_b=*/false, lv,
        /*c_mod=*/(short)0, acc, /*reuse_a=*/false, /*reuse_b=*/false);
  }

  // Every column of D is identical; lanes with li==0 (lanes 0 and 16) hold
  // rows [rowBase + 8h, rowBase + 8h + 8) in acc[0..7].
  if (li == 0) {
    const long base = rowBase + (h << 3);
    float* o = out + base;
    if (base + 8 <= nrows) {
      v4f r0 = { __builtin_amdgcn_exp2f(acc[0]), __builtin_amdgcn_exp2f(acc[1]),
                 __builtin_amdgcn_exp2f(acc[2]), __builtin_amdgcn_exp2f(acc[3]) };
      v4f r1 = { __builtin_amdgcn_exp2f(acc[4]), __builtin_amdgcn_exp2f(acc[5]),
                 __builtin_amdgcn_exp2f(acc[6]), __builtin_amdgcn_exp2f(acc[7]) };
      *(v4f*)(o)     = r0;                 // global_store_b128
      *(v4f*)(o + 4) = r1;                 // global_store_b128
    } else {
#pragma unroll
      for (int v = 0; v < 8; ++v)
        if (base + v < nrows) o[v] = __builtin_amdgcn_exp2f(acc[v]);
    }
  }
}

extern "C" void kernel_launch(void* const* d_in, const int* in_sizes, int n_in,
                              void* d_out, int out_size, void* d_ws, size_t ws_size,
                              hipStream_t stream) {
  const float* x   = (const float*)d_in[0];   // [N, 64] float32
  const float* idw = (const float*)d_in[1];   // [64]    float32
  const float* lnw = (const float*)d_in[2];   // [1, 64] float32
  float*       out = (float*)d_out;           // [N, 1]  float32

  const long nrows  = (long)out_size;         // N
  const int  ntiles = (int)((nrows + 15) / 16);
  const int  waves_per_block = 8;             // 256 threads = 8 wave32
  const int  blocks = (ntiles + waves_per_block - 1) / waves_per_block;

  pta_log_wmma_kernel<<<blocks, 256, 0, stream>>>(x, idw, lnw, out, ntiles, nrows);
}